// TopKSoftmax_755914244779
// MI455X (gfx1250) — compile-verified
//
#include <hip/hip_runtime.h>

// TopK smoothed softmax over the last axis.
// Shapes (from reference): logits (2,8,2048,2048) f32, softmax dim = last, k = 10.
// One workgroup (256 threads = 8 wave32) per row of 2048 floats, 8 elems/thread.
// Row is staged global->LDS with CDNA5 async-to-LDS loads (ASYNCcnt path),
// with non-temporal hints since every byte is touched exactly once (streaming,
// working set 512 MiB >> 192 MB L2).

#define ROW_N 2048
#define TOPK  10
#define TPB   256
#define EPT   8           // elements per thread
#define NWAVE (TPB / 32)

typedef float v4f __attribute__((ext_vector_type(4)));

// Monotonic float<->uint mapping (total order matching float compare, no NaNs in input).
__device__ __forceinline__ unsigned f2mono(float f) {
    unsigned u = __float_as_uint(f);
    return (u & 0x80000000u) ? ~u : (u | 0x80000000u);
}
__device__ __forceinline__ float mono2f(unsigned m) {
    unsigned u = (m & 0x80000000u) ? (m & 0x7fffffffu) : ~m;
    return __uint_as_float(u);
}

__device__ __forceinline__ unsigned long long shfl_xor_u64(unsigned long long x, int mask) {
    unsigned lo = (unsigned)__shfl_xor((int)(unsigned)x, mask, 32);
    unsigned hi = (unsigned)__shfl_xor((int)(unsigned)(x >> 32), mask, 32);
    return ((unsigned long long)hi << 32) | (unsigned long long)lo;
}

__global__ __launch_bounds__(TPB) void topk_smooth_softmax_kernel(
        const float* __restrict__ in, float* __restrict__ out) {
    __shared__ __align__(16) float srow[ROW_N];
    __shared__ unsigned long long redk[NWAVE];
    __shared__ float              redf[NWAVE];

    const int tid  = threadIdx.x;
    const int lane = tid & 31;
    const int wv   = tid >> 5;
    const long long row = blockIdx.x;

    // ---- Stage this row into LDS via CDNA5 async global->LDS loads (NT: stream) ----
    // Each lane moves its own contiguous 32 bytes (two b128 transfers).
    {
        unsigned long long g = (unsigned long long)(const void*)(in + row * ROW_N + (long long)tid * EPT);
        unsigned l = (unsigned)(unsigned long long)(const void*)&srow[tid * EPT];
        asm volatile("global_load_async_to_lds_b128 %0, %1, off th:TH_LOAD_NT"
                     :: "v"(l), "v"(g) : "memory");
        asm volatile("global_load_async_to_lds_b128 %0, %1, off offset:16 th:TH_LOAD_NT"
                     :: "v"(l), "v"(g) : "memory");
        asm volatile("s_wait_asynccnt 0" ::: "memory");
    }
    __syncthreads();

    // ---- Pull my 8 elements into registers (ds_load_b128 x2) ----
    float v[EPT];
    {
        const float4 a = *(const float4*)&srow[tid * EPT];
        const float4 b = *(const float4*)&srow[tid * EPT + 4];
        v[0] = a.x; v[1] = a.y; v[2] = a.z; v[3] = a.w;
        v[4] = b.x; v[5] = b.y; v[6] = b.z; v[7] = b.w;
    }

    // 64-bit keys: value-monotonic in high 32 bits, (N-1-idx) in low bits so
    // equal values prefer the LOWER index, matching jax.lax.top_k tie-breaking.
    unsigned long long key[EPT];
#pragma unroll
    for (int j = 0; j < EPT; ++j) {
        unsigned idx = (unsigned)(tid * EPT + j);
        key[j] = ((unsigned long long)f2mono(v[j]) << 32) |
                 (unsigned long long)(ROW_N - 1u - idx);
    }

    // ---- Extract the TOPK largest keys (iterated block max) ----
    unsigned long long bound = ~0ull;       // strictly-decreasing extraction bound
    float v0 = 0.0f;                        // row max (first extracted value)
    float sum_top = 0.0f;                   // sum exp(topk - v0)
    for (int i = 0; i < TOPK; ++i) {
        unsigned long long lm = 0ull;
#pragma unroll
        for (int j = 0; j < EPT; ++j)
            if (key[j] < bound && key[j] > lm) lm = key[j];
        // wave32 reduction
#pragma unroll
        for (int off = 16; off >= 1; off >>= 1) {
            unsigned long long o = shfl_xor_u64(lm, off);
            if (o > lm) lm = o;
        }
        if (lane == 0) redk[wv] = lm;
        __syncthreads();
        // Every thread reduces the 8 per-wave maxima itself (broadcast LDS
        // reads, no single-thread serialization, no second broadcast word).
        unsigned long long km = redk[0];
#pragma unroll
        for (int w = 1; w < NWAVE; ++w) {
            unsigned long long o = redk[w];
            if (o > km) km = o;
        }
        __syncthreads();                    // protect redk before next write
        const float val = mono2f((unsigned)(km >> 32));
        if (i == 0) v0 = val;
        sum_top += expf(val - v0);
        bound = km;
    }
    const unsigned long long thr = bound;   // key of the k-th largest element

    // ---- Sum of exp over the remaining (non-top-k) elements ----
    float e[EPT];
    float lsum = 0.0f;
#pragma unroll
    for (int j = 0; j < EPT; ++j) {
        e[j] = expf(v[j] - v0);
        if (key[j] < thr) lsum += e[j];
    }
#pragma unroll
    for (int off = 16; off >= 1; off >>= 1)
        lsum += __shfl_xor(lsum, off, 32);
    if (lane == 0) redf[wv] = lsum;
    __syncthreads();
    float sum_rem = redf[0];
#pragma unroll
    for (int w = 1; w < NWAVE; ++w) sum_rem += redf[w];

    // ---- Blend and store:  topk -> 0.9*softmax(topk), else -> 0.1*softmax(rest) ----
    const float ct = 0.9f / sum_top;
    const float cr = 0.1f / sum_rem;
    float o[EPT];
#pragma unroll
    for (int j = 0; j < EPT; ++j)
        o[j] = e[j] * ((key[j] >= thr) ? ct : cr);

    v4f* op = (v4f*)(out + row * ROW_N + (long long)tid * EPT);
    v4f o0 = { o[0], o[1], o[2], o[3] };
    v4f o1 = { o[4], o[5], o[6], o[7] };
    __builtin_nontemporal_store(o0, op);        // NT store: streamed output
    __builtin_nontemporal_store(o1, op + 1);
}

extern "C" void kernel_launch(void* const* d_in, const int* in_sizes, int n_in,
                              void* d_out, int out_size, void* d_ws, size_t ws_size,
                              hipStream_t stream) {
    (void)n_in; (void)d_ws; (void)ws_size; (void)out_size;
    const float* logits = (const float*)d_in[0];   // (2,8,2048,2048) f32
    float* out = (float*)d_out;
    const int total = in_sizes[0];
    const int rows  = total / ROW_N;               // 32768 rows; dim==3 (last axis)
    topk_smooth_softmax_kernel<<<dim3(rows), dim3(TPB), 0, stream>>>(logits, out);
}